// Encoder_48275432407145
// MI455X (gfx1250) — compile-verified
//
#include <hip/hip_runtime.h>

// CDNA5 / gfx1250, wave32.
typedef __attribute__((ext_vector_type(2))) float v2f;
typedef __attribute__((ext_vector_type(8))) float v8f;

#define FEAT   128
#define KNEI   16
#define TILE_M 16
#define PAD    132   // 132 % 64 == 4 -> conflict-free A-fragment reads
#define BN_EPS 1e-5f
#define SLOPE  0.01f
#define RED_BLOCKS 64

// ---------------------------------------------------------------------------
// Kernel A: fused gather-mean + GEMM tile (WMMA f32 16x16x4) + BN partials.
// One block = 16 FULL batch rows (tail handled by enc_tail); 8 waves each own
// a 16-wide N tile of the output. No bounds checks anywhere.
// ---------------------------------------------------------------------------
__global__ __launch_bounds__(256)
void enc_gather_gemm(const float* __restrict__ raw,
                     const float* __restrict__ weight,
                     const int*   __restrict__ neigh,   // neigh_idx (int32 view)
                     float*       __restrict__ xout,    // [nbatch, 128] (d_out as scratch)
                     float*       __restrict__ psum,    // [npartial, 128]
                     float*       __restrict__ psq)     // [npartial, 128]
{
    __shared__ float sNF[TILE_M][PAD];

    const int tid  = threadIdx.x;
    const int bid  = blockIdx.x;
    const int row0 = bid * TILE_M;

    // ---- gather + mean: 16 threads per row, 8 feats per thread ----
    {
        const int r   = tid >> 4;          // 0..15  (row within tile)
        const int l16 = tid & 15;          // 0..15  (feature group)
        float4 acc0 = make_float4(0.f, 0.f, 0.f, 0.f);
        float4 acc1 = make_float4(0.f, 0.f, 0.f, 0.f);
        const long base = (long)(row0 + r) * KNEI;
        for (int k = 0; k < KNEI; ++k) {
            const int idx = neigh[base + k];
            const float4* rp = (const float4*)(raw + (long)idx * FEAT);
            float4 a = rp[l16 * 2 + 0];
            float4 b = rp[l16 * 2 + 1];
            acc0.x += a.x; acc0.y += a.y; acc0.z += a.z; acc0.w += a.w;
            acc1.x += b.x; acc1.y += b.y; acc1.z += b.z; acc1.w += b.w;
        }
        const float inv = 1.0f / (float)KNEI;
        acc0.x *= inv; acc0.y *= inv; acc0.z *= inv; acc0.w *= inv;
        acc1.x *= inv; acc1.y *= inv; acc1.z *= inv; acc1.w *= inv;
        *(float4*)&sNF[r][l16 * 8 + 0] = acc0;   // 16B-aligned (528B row pitch)
        *(float4*)&sNF[r][l16 * 8 + 4] = acc1;
    }
    __syncthreads();

    // ---- GEMM: wave w computes columns [w*16, w*16+16) for the 16-row tile ----
    const int lane = tid & 31;
    const int wv   = tid >> 5;                 // 0..7
    const int n0   = wv * 16;
    const int m    = lane & 15;                // A row / B-D column (per half-wave)
    const int koff = (lane >> 4) << 1;         // 0 (lanes 0-15) or 2 (lanes 16-31)
    const int n    = n0 + m;

    v8f c = {};
    #pragma unroll 8
    for (int kk = 0; kk < FEAT; kk += 4) {
        v2f a, b;
        // A 16x4 f32 layout: VGPR0 = K=koff, VGPR1 = K=koff+1
        a.x = sNF[m][kk + koff];
        a.y = sNF[m][kk + koff + 1];
        // B 4x16 f32 layout: VGPR0 = row K=koff, VGPR1 = row K=koff+1 (lane = N)
        b.x = weight[(kk + koff)     * FEAT + n];
        b.y = weight[(kk + koff + 1) * FEAT + n];
        c = __builtin_amdgcn_wmma_f32_16x16x4_f32(
                /*neg_a=*/false, a, /*neg_b=*/false, b,
                /*c_mod=*/(short)0, c, /*reuse_a=*/false, /*reuse_b=*/false);
    }

    // ---- write x tile (D layout: VGPR v -> M = v + 8*(lane>=16), N = lane&15) ----
    const int rbase = row0 + ((lane >> 4) << 3);
    float* dst = xout + (long)rbase * FEAT + n;
    #pragma unroll
    for (int v = 0; v < 8; ++v) dst[(long)v * FEAT] = c[v];

    // ---- BN partials: per-column sum & sumsq over this tile's 16 rows ----
    float ps = 0.f, pq = 0.f;
    #pragma unroll
    for (int v = 0; v < 8; ++v) { ps += c[v]; pq += c[v] * c[v]; }
    // lanes L and L+16 hold the same column (rows 0-7 vs 8-15): fold halves
    ps += __shfl_down(ps, 16, 32);
    pq += __shfl_down(pq, 16, 32);
    if (lane < 16) {
        psum[(long)bid * FEAT + n] = ps;
        psq [(long)bid * FEAT + n] = pq;
    }
}

// ---------------------------------------------------------------------------
// Kernel A': scalar tail for nbatch % 16 rows (launched only when needed;
// never launched for the reference shape). Writes one extra partial row.
// ---------------------------------------------------------------------------
__global__ __launch_bounds__(128)
void enc_tail(const float* __restrict__ raw,
              const float* __restrict__ weight,
              const int*   __restrict__ neigh,
              float*       __restrict__ xout,
              float*       __restrict__ psum,   // row `tailRow` of partials
              float*       __restrict__ psq,
              int row_start, int nbatch, int tailRow)
{
    __shared__ float sF[FEAT];
    const int n = threadIdx.x;                 // 0..127 (output column)
    float ps = 0.f, pq = 0.f;
    for (int r = row_start; r < nbatch; ++r) {
        float acc = 0.f;
        const long base = (long)r * KNEI;
        for (int k = 0; k < KNEI; ++k)
            acc += raw[(long)neigh[base + k] * FEAT + n];
        sF[n] = acc * (1.0f / (float)KNEI);
        __syncthreads();
        float x = 0.f;
        for (int f = 0; f < FEAT; ++f) x += sF[f] * weight[f * FEAT + n];
        xout[(long)r * FEAT + n] = x;
        ps += x; pq += x * x;
        __syncthreads();
    }
    psum[(long)tailRow * FEAT + n] = ps;
    psq [(long)tailRow * FEAT + n] = pq;
}

// ---------------------------------------------------------------------------
// Kernel B1: parallel tree reduction of per-block partials (coalesced).
// ---------------------------------------------------------------------------
__global__ __launch_bounds__(128)
void bn_reduce(const float* __restrict__ psum, const float* __restrict__ psq,
               float* __restrict__ s2sum, float* __restrict__ s2sq, int nrows)
{
    const int nI    = threadIdx.x;                       // 0..127 (column)
    const int blk   = blockIdx.x;                        // 0..RED_BLOCKS-1
    const int chunk = (nrows + RED_BLOCKS - 1) / RED_BLOCKS;
    const int start = blk * chunk;
    const int end   = (start + chunk < nrows) ? (start + chunk) : nrows;
    float s = 0.f, q = 0.f;
    for (int b = start; b < end; ++b) {
        s += psum[(long)b * FEAT + nI];
        q += psq [(long)b * FEAT + nI];
    }
    s2sum[(long)blk * FEAT + nI] = s;
    s2sq [(long)blk * FEAT + nI] = q;
}

// ---------------------------------------------------------------------------
// Kernel B2: finalize -> scale/shift per feature column.
// ---------------------------------------------------------------------------
__global__ __launch_bounds__(128)
void bn_finalize(const float* __restrict__ s2sum, const float* __restrict__ s2sq,
                 const float* __restrict__ gamma, const float* __restrict__ beta,
                 float* __restrict__ scale, float* __restrict__ shift, float invN)
{
    const int nI = threadIdx.x;   // 0..127
    float s = 0.f, q = 0.f;
    #pragma unroll 8
    for (int b = 0; b < RED_BLOCKS; ++b) {
        s += s2sum[(long)b * FEAT + nI];
        q += s2sq [(long)b * FEAT + nI];
    }
    const float mean = s * invN;
    const float var  = q * invN - mean * mean;
    const float sc   = gamma[nI] * rsqrtf(var + BN_EPS);
    scale[nI] = sc;
    shift[nI] = beta[nI] - mean * sc;
}

// ---------------------------------------------------------------------------
// Kernel C: in-place normalize + LeakyReLU, float4-vectorized.
// ---------------------------------------------------------------------------
__global__ __launch_bounds__(256)
void bn_apply(float* __restrict__ x, const float* __restrict__ scale,
              const float* __restrict__ shift, long total4)
{
    const long i = (long)blockIdx.x * blockDim.x + threadIdx.x;
    if (i >= total4) return;
    const int nI = (int)((i * 4) & (FEAT - 1));
    float4 v  = ((float4*)x)[i];
    const float4 sc = *(const float4*)(scale + nI);
    const float4 sh = *(const float4*)(shift + nI);
    v.x = v.x * sc.x + sh.x;
    v.y = v.y * sc.y + sh.y;
    v.z = v.z * sc.z + sh.z;
    v.w = v.w * sc.w + sh.w;
    v.x = (v.x >= 0.f) ? v.x : SLOPE * v.x;
    v.y = (v.y >= 0.f) ? v.y : SLOPE * v.y;
    v.z = (v.z >= 0.f) ? v.z : SLOPE * v.z;
    v.w = (v.w >= 0.f) ? v.w : SLOPE * v.w;
    ((float4*)x)[i] = v;
}

// ---------------------------------------------------------------------------
extern "C" void kernel_launch(void* const* d_in, const int* in_sizes, int n_in,
                              void* d_out, int out_size, void* d_ws, size_t ws_size,
                              hipStream_t stream)
{
    (void)n_in; (void)out_size; (void)ws_size;

    const float* raw    = (const float*)d_in[0];   // [100000,128] f32
    const float* weight = (const float*)d_in[1];   // [128,128]    f32
    const float* gamma  = (const float*)d_in[2];   // [128]        f32
    const float* beta   = (const float*)d_in[3];   // [128]        f32
    const int*   neigh  = (const int*)  d_in[4];   // [nbatch,16]  int

    float* out = (float*)d_out;                    // [nbatch,128] f32 (also GEMM scratch)
    float* ws  = (float*)d_ws;

    const int nbatch   = in_sizes[4] / KNEI;       // 50000
    const int nfull    = nbatch / TILE_M;          // 3125 full tiles
    const int tail     = nbatch - nfull * TILE_M;  // 0 for the reference shape
    const int npartial = nfull + (tail ? 1 : 0);

    float* psum  = ws;                                    // npartial*128 floats
    float* psq   = psum  + (long)npartial * FEAT;         // npartial*128 floats
    float* s2sum = psq   + (long)npartial * FEAT;         // RED_BLOCKS*128 floats
    float* s2sq  = s2sum + (long)RED_BLOCKS * FEAT;       // RED_BLOCKS*128 floats
    float* scale = s2sq  + (long)RED_BLOCKS * FEAT;       // 128 floats
    float* shift = scale + FEAT;                          // 128 floats

    if (nfull > 0)
        enc_gather_gemm<<<nfull, 256, 0, stream>>>(raw, weight, neigh, out,
                                                   psum, psq);
    if (tail > 0)
        enc_tail<<<1, 128, 0, stream>>>(raw, weight, neigh, out, psum, psq,
                                        nfull * TILE_M, nbatch, nfull);

    bn_reduce<<<RED_BLOCKS, 128, 0, stream>>>(psum, psq, s2sum, s2sq, npartial);
    bn_finalize<<<1, 128, 0, stream>>>(s2sum, s2sq, gamma, beta, scale, shift,
                                       1.0f / (float)nbatch);
    const long total4 = (long)nbatch * FEAT / 4;
    bn_apply<<<(int)((total4 + 255) / 256), 256, 0, stream>>>(out, scale, shift, total4);
}